// model_joint_51642686767181
// MI455X (gfx1250) — compile-verified
//
#include <hip/hip_runtime.h>

// ---------------------------------------------------------------------------
// CDNA5 (gfx1250) fp32 WMMA GEMM pipeline for triple-AE + recurrent mixer.
// Wave32; 8 waves/block; block tile 128x64; wave tile 32x32 (4 accumulators).
// Double-buffered async global->LDS copies (ASYNCcnt) pipelined against WMMA.
// ---------------------------------------------------------------------------

typedef __attribute__((ext_vector_type(2))) float v2f;
typedef __attribute__((ext_vector_type(8))) float v8f;

#define TM 128         // block tile M
#define TN 64          // block tile N
#define KC 32          // K chunk staged in LDS
#define APAD 36        // padded LDS row for A (conflict-free, 16B aligned)
#define BPAD 72        // padded LDS row for B (16B aligned)

#if __has_builtin(__builtin_amdgcn_global_load_async_to_lds_b128)
#define HAVE_ASYNC_LDS 1
#else
#define HAVE_ASYNC_LDS 0
#endif

#if HAVE_ASYNC_LDS
// Builtin prototype (from clang diagnostic): param0 = int4 in global AS,
// param1 = int4 in LDS AS, then two immediate ints (offset, cpol).
typedef __attribute__((__vector_size__(4 * sizeof(int)))) int v4i;
typedef __attribute__((address_space(1))) v4i gv4i;
typedef __attribute__((address_space(3))) v4i lv4i;
#endif

// C[M,N] = act(A[M,K] @ W[K,N] + bias[N]); row-major, lda/ldc strides.
__global__ __launch_bounds__(256) void gemm_bias_act(
    const float* __restrict__ A, int lda,
    const float* __restrict__ W,          // K x N, stride N
    const float* __restrict__ bias,
    float* __restrict__ C, int ldc,
    int M, int N, int K, int relu)
{
    __shared__ float sA[2][TM][APAD];     // double-buffered 128 x 32 (padded)
    __shared__ float sB[2][KC][BPAD];     // double-buffered 32 x 64 (padded)

    const int tid  = threadIdx.x;
    const int lane = tid & 31;
    const int wave = tid >> 5;            // 0..7
    const int wm   = wave >> 1;           // 0..3 -> 32-row sub-tile
    const int wn   = wave & 1;            // 0..1 -> 32-col sub-tile
    const int l16  = lane & 15;
    const int hi   = lane >> 4;           // 0 or 1
    const int khalf = hi * 2;             // lanes 16-31 hold K=2,3 slice

    const int blockM = blockIdx.x * TM;
    const int blockN = blockIdx.y * TN;

    // A loaders: 128x32 = 1024 float4, 4 per thread
    const int arow = tid >> 3;            // 0..31 (+32 per group)
    const int ac4  = (tid & 7) << 2;
    // B loaders: 32x64 = 512 float4, 2 per thread
    const int brow = tid >> 4;            // 0..15 (+16 per group)
    const int bc4  = (tid & 15) << 2;

    // Issue one K-stage of global->LDS staging into buffer `buf`.
    auto issue_stage = [&](int kk, int buf) {
        const float* a0 = A + (size_t)(blockM + arow) * lda + kk + ac4;
        const float* a1 = a0 + (size_t)32 * lda;
        const float* a2 = a0 + (size_t)64 * lda;
        const float* a3 = a0 + (size_t)96 * lda;
        const float* b0 = W + (size_t)(kk + brow) * N + blockN + bc4;
        const float* b1 = b0 + (size_t)16 * N;
#if HAVE_ASYNC_LDS
        // gfx1250 async global->LDS (ASYNCcnt-tracked, no VGPR staging)
        __builtin_amdgcn_global_load_async_to_lds_b128(
            (gv4i*)a0, (lv4i*)&sA[buf][arow][ac4], 0, 0);
        __builtin_amdgcn_global_load_async_to_lds_b128(
            (gv4i*)a1, (lv4i*)&sA[buf][arow + 32][ac4], 0, 0);
        __builtin_amdgcn_global_load_async_to_lds_b128(
            (gv4i*)a2, (lv4i*)&sA[buf][arow + 64][ac4], 0, 0);
        __builtin_amdgcn_global_load_async_to_lds_b128(
            (gv4i*)a3, (lv4i*)&sA[buf][arow + 96][ac4], 0, 0);
        __builtin_amdgcn_global_load_async_to_lds_b128(
            (gv4i*)b0, (lv4i*)&sB[buf][brow][bc4], 0, 0);
        __builtin_amdgcn_global_load_async_to_lds_b128(
            (gv4i*)b1, (lv4i*)&sB[buf][brow + 16][bc4], 0, 0);
#else
        float4 ra0 = *(const float4*)a0;
        float4 ra1 = *(const float4*)a1;
        float4 ra2 = *(const float4*)a2;
        float4 ra3 = *(const float4*)a3;
        float4 rb0 = *(const float4*)b0;
        float4 rb1 = *(const float4*)b1;
        *(float4*)&sA[buf][arow][ac4]       = ra0;
        *(float4*)&sA[buf][arow + 32][ac4]  = ra1;
        *(float4*)&sA[buf][arow + 64][ac4]  = ra2;
        *(float4*)&sA[buf][arow + 96][ac4]  = ra3;
        *(float4*)&sB[buf][brow][bc4]       = rb0;
        *(float4*)&sB[buf][brow + 16][bc4]  = rb1;
        if (kk + KC < K) {                 // gfx1250 global_prefetch_b8
            __builtin_prefetch(a0 + KC, 0, 0);
            __builtin_prefetch(b0 + (size_t)KC * N, 0, 0);
        }
#endif
    };

    v8f acc00 = {}, acc01 = {}, acc10 = {}, acc11 = {};

    issue_stage(0, 0);                     // prologue
    int p = 0;
    for (int kk = 0; kk < K; kk += KC) {
        const bool more = (kk + KC) < K;
        if (more)
            issue_stage(kk + KC, p ^ 1);   // overlap next stage's DMA
#if HAVE_ASYNC_LDS
        if (more)
            asm volatile("s_wait_asynccnt 6" ::: "memory");  // this stage done
        else
            asm volatile("s_wait_asynccnt 0" ::: "memory");
#endif
        __syncthreads();

        // ---- 32x V_WMMA_F32_16X16X4_F32 over the KC chunk ----
        #pragma unroll
        for (int k4 = 0; k4 < KC; k4 += 4) {
            v2f av0, av1, bv0, bv1;
            // A 16x4 fragment: lanes 0-15 -> K=k4+0,1 ; lanes 16-31 -> K=k4+2,3
            av0.x = sA[p][wm * 32 + l16][k4 + khalf];
            av0.y = sA[p][wm * 32 + l16][k4 + khalf + 1];
            av1.x = sA[p][wm * 32 + 16 + l16][k4 + khalf];
            av1.y = sA[p][wm * 32 + 16 + l16][k4 + khalf + 1];
            // B 4x16 fragment: rows striped across lanes, K halves per lane half
            bv0.x = sB[p][k4 + khalf][wn * 32 + l16];
            bv0.y = sB[p][k4 + khalf + 1][wn * 32 + l16];
            bv1.x = sB[p][k4 + khalf][wn * 32 + 16 + l16];
            bv1.y = sB[p][k4 + khalf + 1][wn * 32 + 16 + l16];
            acc00 = __builtin_amdgcn_wmma_f32_16x16x4_f32(
                false, av0, false, bv0, (short)0, acc00, false, false);
            acc01 = __builtin_amdgcn_wmma_f32_16x16x4_f32(
                false, av0, false, bv1, (short)0, acc01, false, false);
            acc10 = __builtin_amdgcn_wmma_f32_16x16x4_f32(
                false, av1, false, bv0, (short)0, acc10, false, false);
            acc11 = __builtin_amdgcn_wmma_f32_16x16x4_f32(
                false, av1, false, bv1, (short)0, acc11, false, false);
        }
        __syncthreads();                   // buffer p free for stage kk+2*KC
        p ^= 1;
    }

    // ---- epilogue: bias + optional ReLU; C/D VGPR r -> rows r / r+8 ----
    const int col0 = blockN + wn * 32 + l16;
    const int col1 = col0 + 16;
    const float bb0 = bias ? bias[col0] : 0.0f;
    const float bb1 = bias ? bias[col1] : 0.0f;
    const int rbase0 = blockM + wm * 32 + 8 * hi;
    #pragma unroll
    for (int r = 0; r < 8; ++r) {
        const int row0 = rbase0 + r;
        const int row1 = row0 + 16;
        float v00 = acc00[r] + bb0;
        float v01 = acc01[r] + bb1;
        float v10 = acc10[r] + bb0;
        float v11 = acc11[r] + bb1;
        if (relu) {
            v00 = fmaxf(v00, 0.0f); v01 = fmaxf(v01, 0.0f);
            v10 = fmaxf(v10, 0.0f); v11 = fmaxf(v11, 0.0f);
        }
        C[(size_t)row0 * ldc + col0] = v00;
        C[(size_t)row0 * ldc + col1] = v01;
        C[(size_t)row1 * ldc + col0] = v10;
        C[(size_t)row1 * ldc + col1] = v11;
    }
}

// u_y[:,0,:] = en0[:,0,:]   (strided rows, stride 96*64)
__global__ void copy_u0_kernel(const float* __restrict__ en0,
                               float* __restrict__ uy)
{
    const int s = blockIdx.x;       // 0..511
    const int z = threadIdx.x;      // 0..63
    uy[(size_t)s * 96 * 64 + z] = en0[(size_t)s * 96 * 64 + z];
}

// CAT[s, 0:64]=u_prev[s], CAT[s,64:128]=e1[s], CAT[s,128:192]=e2[s]
__global__ void concat_kernel(const float* __restrict__ u_prev,
                              const float* __restrict__ e1t,
                              const float* __restrict__ e2t,
                              float* __restrict__ cat)
{
    const int s = blockIdx.x;       // 0..511
    const int z = threadIdx.x;      // 0..63
    const size_t rs = (size_t)s * 96 * 64;
    cat[s * 192 + z]       = u_prev[rs + z];
    cat[s * 192 + 64 + z]  = e1t[rs + z];
    cat[s * 192 + 128 + z] = e2t[rs + z];
}

// ---------------------------------------------------------------------------
// Host side
// ---------------------------------------------------------------------------

#define ROWS   49152
#define D_IN   1024
#define D_Z    64
#define NSAMP  512
#define TSTEP  96
#define CHUNK  8192

struct LinP { const float* W; const float* b; };

extern "C" void kernel_launch(void* const* d_in, const int* in_sizes, int n_in,
                              void* d_out, int out_size, void* d_ws, size_t ws_size,
                              hipStream_t stream)
{
    (void)out_size; (void)ws_size;

    // ---- classify inputs by flat element count (robust to pytree ordering) ----
    const float* X[3] = {nullptr, nullptr, nullptr};
    LinP enc[3][4], dec[3][4], mm[3];
    bool used[128] = {false};

    int nx = 0;
    for (int i = 0; i < n_in; ++i) {
        if (in_sizes[i] == ROWS * D_IN && nx < 3) {     // 50,331,648
            X[nx++] = (const float*)d_in[i];
            used[i] = true;
        }
    }
    for (int i = 0; i < n_in; ++i) {
        if (in_sizes[i] == 3 * D_Z * 512) {             // 98,304 : mm.W0 (unique)
            for (int l = 0; l < 3; ++l) {
                mm[l].W = (const float*)d_in[i + 2 * l];
                mm[l].b = (const float*)d_in[i + 2 * l + 1];
            }
            for (int j = 0; j < 6; ++j) used[i + j] = true;
            break;
        }
    }
    int rem[48], r = 0;
    for (int i = 0; i < n_in && r < 48; ++i)
        if (!used[i]) rem[r++] = i;
    for (int a = 0; a < 3; ++a) {
        const int* blk = rem + a * 16;
        const bool encFirst = (in_sizes[blk[0]] == 1024 * 512); // enc.W0 first?
        const int eo = encFirst ? 0 : 8;
        const int dofs = encFirst ? 8 : 0;
        for (int l = 0; l < 4; ++l) {
            enc[a][l].W = (const float*)d_in[blk[eo + 2 * l]];
            enc[a][l].b = (const float*)d_in[blk[eo + 2 * l + 1]];
            dec[a][l].W = (const float*)d_in[blk[dofs + 2 * l]];
            dec[a][l].b = (const float*)d_in[blk[dofs + 2 * l + 1]];
        }
    }

    // ---- output layout (flat concat in return order) ----
    float* out = (float*)d_out;
    const size_t SZ_EN = (size_t)ROWS * D_Z;     // 3,145,728
    const size_t SZ_DE = (size_t)ROWS * D_IN;    // 50,331,648
    float* EN[3] = { out, out + SZ_EN, out + 2 * SZ_EN };
    float* DE[3] = { out + 3 * SZ_EN,
                     out + 3 * SZ_EN + SZ_DE,
                     out + 3 * SZ_EN + 2 * SZ_DE };
    float* UY   = out + 3 * SZ_EN + 3 * SZ_DE;
    float* UYDE = UY + SZ_EN;

    // ---- workspace carve (~28 MB) ----
    float* ws = (float*)d_ws;
    float* S1  = ws;                               // 8192 x 512
    float* S2  = S1 + (size_t)CHUNK * 512;         // 8192 x 256
    float* CAT = S2 + (size_t)CHUNK * 256;         // 512 x 192
    float* H1  = CAT + (size_t)NSAMP * 192;        // 512 x 512
    float* H2  = H1 + (size_t)NSAMP * 512;         // 512 x 512

    auto gemm = [&](const float* A, int lda, const LinP& p,
                    float* C, int ldc, int M, int N, int K, int relu) {
        dim3 grid(M / TM, N / TN);
        gemm_bias_act<<<grid, 256, 0, stream>>>(A, lda, p.W, p.b, C, ldc,
                                                M, N, K, relu);
    };

    // ---- autoencoders: enc + dec, chunked over rows ----
    for (int a = 0; a < 3; ++a) {
        for (int c = 0; c < ROWS / CHUNK; ++c) {
            const float* xa = X[a] + (size_t)c * CHUNK * D_IN;
            float* en = EN[a] + (size_t)c * CHUNK * D_Z;
            float* de = DE[a] + (size_t)c * CHUNK * D_IN;
            gemm(xa, 1024, enc[a][0], S1, 512, CHUNK, 512, 1024, 1);
            gemm(S1, 512,  enc[a][1], S2, 256, CHUNK, 256, 512,  1);
            gemm(S2, 256,  enc[a][2], S1, 128, CHUNK, 128, 256,  1);
            gemm(S1, 128,  enc[a][3], en, 64,  CHUNK, 64,  128,  0);
            gemm(en, 64,   dec[a][0], S1, 128, CHUNK, 128, 64,   1);
            gemm(S1, 128,  dec[a][1], S2, 256, CHUNK, 256, 128,  1);
            gemm(S2, 256,  dec[a][2], S1, 512, CHUNK, 512, 256,  1);
            gemm(S1, 512,  dec[a][3], de, 1024, CHUNK, 1024, 512, 0);
        }
    }

    // ---- recurrent mixer: u_t = mm(concat[u_{t-1}, e1_{t-1}, e2_{t-1}]) ----
    copy_u0_kernel<<<NSAMP, D_Z, 0, stream>>>(EN[0], UY);
    for (int t = 1; t < TSTEP; ++t) {
        concat_kernel<<<NSAMP, D_Z, 0, stream>>>(UY + (size_t)(t - 1) * D_Z,
                                                 EN[1] + (size_t)(t - 1) * D_Z,
                                                 EN[2] + (size_t)(t - 1) * D_Z,
                                                 CAT);
        gemm(CAT, 192, mm[0], H1, 512, NSAMP, 512, 192, 1);
        gemm(H1, 512,  mm[1], H2, 512, NSAMP, 512, 512, 1);
        gemm(H2, 512,  mm[2], UY + (size_t)t * D_Z, TSTEP * D_Z,
             NSAMP, 64, 512, 0);
    }

    // ---- u_y_de = dec0(u_y), chunked ----
    for (int c = 0; c < ROWS / CHUNK; ++c) {
        const float* uy = UY + (size_t)c * CHUNK * D_Z;
        float* de = UYDE + (size_t)c * CHUNK * D_IN;
        gemm(uy, 64,  dec[0][0], S1, 128, CHUNK, 128, 64,  1);
        gemm(S1, 128, dec[0][1], S2, 256, CHUNK, 256, 128, 1);
        gemm(S2, 256, dec[0][2], S1, 512, CHUNK, 512, 256, 1);
        gemm(S1, 512, dec[0][3], de, 1024, CHUNK, 1024, 512, 0);
    }
}